// LAFLayer_15015205667103
// MI455X (gfx1250) — compile-verified
//
#include <hip/hip_runtime.h>
#include <hip/hip_bf16.h>

// LAF layer for MI455X (gfx1250, wave32).
// data:    [N, 32] f32      (d_in[0])
// index:   [N] int (sorted) (d_in[1])
// weights: [12, 4] f32      (d_in[2])
// num_segments scalar       (d_in[3], value derived from out_size instead)
// out:     [S, 32, 4] f32
// scratch: scatter accumulator [S, 32, 4, 4] f32 in d_ws (20.48 MB for S=10000)

#define LAF_EPS 1e-7f
#define LAF_SUP (1.0f - 1e-7f)
#define CHUNK 256          // nodes per block (32 KB data tile)
#define BLOCK_A 128        // 4 waves; lane = f (0..31), wave = u (0..3)

#if defined(__gfx1250__) && __has_builtin(__builtin_amdgcn_tensor_load_to_lds) && \
    __has_builtin(__builtin_amdgcn_s_wait_tensorcnt)
#define LAF_USE_TDM 1
#else
#define LAF_USE_TDM 0
#endif

__device__ __forceinline__ float fast_log2(float x) {
    return __builtin_amdgcn_logf(x);        // v_log_f32 (base-2)
}
__device__ __forceinline__ float fast_exp2(float x) {
    return __builtin_amdgcn_exp2f(x);       // v_exp_f32 (base-2)
}

#if LAF_USE_TDM
typedef unsigned int v4u __attribute__((ext_vector_type(4)));
typedef int          v8i __attribute__((ext_vector_type(8)));
typedef int          v4i __attribute__((ext_vector_type(4)));

// Issue a TDM load of a [rows x 32] f32 tile (row-major, stride 32) into LDS.
// D# per CDNA5 ISA ch.8: group0 = {count/lds_addr/global_addr/type},
// group1 = {data_size, tensor dims/strides, tile dims}. 2D tensor -> remaining groups zero.
// This toolchain's builtin is the 6-arg form: (v4u, v8i, v4i, v4i, v8i, i32 cpol).
__device__ __forceinline__ void tdm_load_rows_to_lds(const float* gsrc, void* lds_dst, int rows) {
    const unsigned           lds = (unsigned)(size_t)lds_dst;   // ISA 10.2: LDS aperture addr[31:0]
    const unsigned long long ga  = (unsigned long long)(size_t)gsrc;
    const unsigned r = (unsigned)rows;

    v4u g0;
    g0.x = 1u;                                                  // count=1, user descriptor
    g0.y = lds;                                                 // lds_addr (bytes)
    g0.z = (unsigned)(ga & 0xFFFFFFFFu);                        // global_addr[31:0]
    g0.w = (unsigned)((ga >> 32) & 0x01FFFFFFu) | (2u << 30);   // global_addr[56:32] | type=2

    v8i g1;
    g1[0] = (int)(2u << 16);                                    // data_size=2 (4 bytes); mask/pad=0
    g1[1] = (int)(32u << 16);                                   // tensor_dim0[15:0]=32 in bits[63:48]
    g1[2] = (int)((r & 0xFFFFu) << 16);                         // tensor_dim0 hi=0 | tensor_dim1 lo=rows
    g1[3] = (int)(((r >> 16) & 0xFFFFu) | (32u << 16));         // tensor_dim1 hi | tile_dim0=32
    g1[4] = (int)(r & 0xFFFFu);                                 // tile_dim1=rows, tile_dim2=0
    g1[5] = 32;                                                 // tensor_dim0_stride=32 (low 32b)
    g1[6] = 0;                                                  // stride0 hi | stride1 lo
    g1[7] = 0;                                                  // stride1 hi

    v4i z4 = {};
    v8i z8 = {};
    __builtin_amdgcn_tensor_load_to_lds(g0, g1, z4, z4, z8, 0);
}
#endif

__global__ void laf_zero_kernel(float* __restrict__ p, int n) {
    int t = blockIdx.x * blockDim.x + threadIdx.x;
    if (t < n) p[t] = 0.0f;
}

// Phase A: pre-scatter pow + segmented sum exploiting sorted index.
// threadIdx.x & 31  -> f  (feature, conflict-free LDS banks / coalesced loads)
// threadIdx.x >> 5  -> u  (wave-uniform -> weights become scalar regs)
__global__ void __launch_bounds__(BLOCK_A)
laf_scatter_kernel(const float* __restrict__ data,
                   const int*   __restrict__ index,
                   const float* __restrict__ weights,
                   float*       __restrict__ scatter,
                   int N) {
    __shared__ int sidx[CHUNK];
#if LAF_USE_TDM
    __shared__ float drows[CHUNK * 32];      // 32 KB staged data tile
#endif

    const int n0  = blockIdx.x * CHUNK;
    const int cnt = min(CHUNK, N - n0);

#if LAF_USE_TDM
    // Wave 0 drives the Tensor Data Mover for this block's tile.
    if (threadIdx.x < 32) {
        tdm_load_rows_to_lds(data + (size_t)n0 * 32, drows, cnt);
    }
#endif

    // Stage this block's index run into LDS (small, plain loads).
    for (int i = threadIdx.x; i < cnt; i += BLOCK_A)
        sidx[i] = index[n0 + i];

#if LAF_USE_TDM
    if (threadIdx.x < 32) {
        __builtin_amdgcn_s_wait_tensorcnt(0);    // drain TDM before publishing LDS
    }
#endif
    __syncthreads();

    const int f = threadIdx.x & 31;   // lane id within wave
    const int u = threadIdx.x >> 5;   // wave id (uniform)

    // relu'd exponent weights for the 4 parity terms (wave-uniform scalars)
    const float w0 = fmaxf(weights[0 * 4 + u], 0.0f);  // base x
    const float w1 = fmaxf(weights[1 * 4 + u], 0.0f);  // base 1-x
    const float w2 = fmaxf(weights[2 * 4 + u], 0.0f);  // base x
    const float w3 = fmaxf(weights[3 * 4 + u], 0.0f);  // base 1-x

    float a0 = 0.0f, a1 = 0.0f, a2 = 0.0f, a3 = 0.0f;
    int scur = sidx[0];

    for (int i = 0; i < cnt; ++i) {
        const int s = sidx[i];               // uniform across the whole block
        if (s != scur) {                     // scalar branch: segment boundary
            const int base = scur * 512 + f * 16 + u;   // ((s*32+f)*4 + j)*4 + u
            __hip_atomic_fetch_add(&scatter[base +  0], a0, __ATOMIC_RELAXED, __HIP_MEMORY_SCOPE_AGENT);
            __hip_atomic_fetch_add(&scatter[base +  4], a1, __ATOMIC_RELAXED, __HIP_MEMORY_SCOPE_AGENT);
            __hip_atomic_fetch_add(&scatter[base +  8], a2, __ATOMIC_RELAXED, __HIP_MEMORY_SCOPE_AGENT);
            __hip_atomic_fetch_add(&scatter[base + 12], a3, __ATOMIC_RELAXED, __HIP_MEMORY_SCOPE_AGENT);
            a0 = a1 = a2 = a3 = 0.0f;
            scur = s;
        }

#if LAF_USE_TDM
        float x = drows[i * 32 + f];                         // ds_load_b32, bank = f
#else
        if (n0 + i + 16 < N)                                 // gfx1250 global_prefetch_b8
            __builtin_prefetch(&data[(size_t)(n0 + i + 16) * 32 + f], 0, 0);
        float x = data[(size_t)(n0 + i) * 32 + f];
#endif
        x = fminf(fmaxf(x, LAF_EPS), LAF_SUP);
        const float lx  = fast_log2(x);           // shared by j=0,2
        const float l1x = fast_log2(1.0f - x);    // shared by j=1,3
        a0 += fast_exp2(w0 * lx);                 // x^w0
        a1 += fast_exp2(w1 * l1x);                // (1-x)^w1
        a2 += fast_exp2(w2 * lx);                 // x^w2
        a3 += fast_exp2(w3 * l1x);                // (1-x)^w3
    }

    // Final flush of the last run.
    const int base = scur * 512 + f * 16 + u;
    __hip_atomic_fetch_add(&scatter[base +  0], a0, __ATOMIC_RELAXED, __HIP_MEMORY_SCOPE_AGENT);
    __hip_atomic_fetch_add(&scatter[base +  4], a1, __ATOMIC_RELAXED, __HIP_MEMORY_SCOPE_AGENT);
    __hip_atomic_fetch_add(&scatter[base +  8], a2, __ATOMIC_RELAXED, __HIP_MEMORY_SCOPE_AGENT);
    __hip_atomic_fetch_add(&scatter[base + 12], a3, __ATOMIC_RELAXED, __HIP_MEMORY_SCOPE_AGENT);
}

// Phase B: clip -> pow(relu(w[4:8])) -> scale(w[8:12]) -> num/den with eps guard.
// One thread per output element t = (s*32+f)*4 + u.
__global__ void laf_finalize_kernel(const float* __restrict__ scatter,
                                    const float* __restrict__ weights,
                                    float*       __restrict__ out,
                                    int total) {
    const int t = blockIdx.x * blockDim.x + threadIdx.x;
    if (t >= total) return;

    const int u    = t & 3;
    const int base = (t >> 2) * 16 + u;   // scatter index for j=0

    float num = 0.0f, den = 0.0f;
#pragma unroll
    for (int j = 0; j < 4; ++j) {
        const float sc = fmaxf(scatter[base + 4 * j], LAF_EPS);
        const float ew = fmaxf(weights[(4 + j) * 4 + u], 0.0f);
        const float ab = weights[(8 + j) * 4 + u];
        const float term = fast_exp2(ew * fast_log2(sc)) * ab;
        if (j < 2) num += term; else den += term;
    }

    // multiplier = 2*max(sign(den),0)-1 : +1 if den>0, else -1 (den==0 -> -1)
    const float mult = (den > 0.0f) ? 1.0f : -1.0f;
    if (den < LAF_EPS && den > -LAF_EPS) den = mult * LAF_EPS;
    out[t] = num / den;
}

extern "C" void kernel_launch(void* const* d_in, const int* in_sizes, int n_in,
                              void* d_out, int out_size, void* d_ws, size_t ws_size,
                              hipStream_t stream) {
    const float* data    = (const float*)d_in[0];
    const int*   index   = (const int*)d_in[1];
    const float* weights = (const float*)d_in[2];
    float*       out     = (float*)d_out;
    float*       scatter = (float*)d_ws;

    const int N     = in_sizes[0] / 32;      // 200000
    const int total = out_size;              // S*32*4
    const int S     = out_size / 128;        // 10000
    const int scatN = S * 512;               // S*32*4*4 accumulator elements

    laf_zero_kernel<<<(scatN + 255) / 256, 256, 0, stream>>>(scatter, scatN);

    const int nblkA = (N + CHUNK - 1) / CHUNK;
    laf_scatter_kernel<<<nblkA, BLOCK_A, 0, stream>>>(data, index, weights, scatter, N);

    laf_finalize_kernel<<<(total + 255) / 256, 256, 0, stream>>>(scatter, weights, out, total);
}